// myBadTransfomerBlock_56495999811681
// MI455X (gfx1250) — compile-verified
//
#include <hip/hip_runtime.h>

typedef __attribute__((ext_vector_type(2))) float v2f;
typedef __attribute__((ext_vector_type(8))) float v8f;

constexpr int   kBATCH = 4;
constexpr int   kN     = 4096;
constexpr int   kD     = 64;
constexpr int   kP     = 66;     // padded LDS row stride (floats): conflict-free b64 frag reads
constexpr float kEPS   = 1e-6f;
constexpr int   kWAVES = 8;      // waves per block (wave32)
constexpr int   kTPB   = kWAVES * 32;            // 256
constexpr int   kTokPerBlk = kWAVES * 16;        // 128 tokens / block
constexpr int   kBlocks    = (kBATCH * kN) / kTokPerBlk;  // 128

// D = A(16x4) * B(4x16) + C, fp32 in / fp32 accum
__device__ __forceinline__ v8f wmma_f32(v2f a, v2f b, v8f c) {
  return __builtin_amdgcn_wmma_f32_16x16x4_f32(false, a, false, b, (short)0, c,
                                               false, false);
}

// One wave computes a 16x64 tile:  D(16x64) = A(16x64) * B(64x64) + D
// arow : per-lane pointer to this lane's A row (row = lane&15), K contiguous.
//        A-frag layout: lanes 0-15 hold K=k0,k0+1 ; lanes 16-31 hold K=k0+2,k0+3
//        -> a single 8-byte load at (k0 + koff).
// bmat : row-major [64][bstride] holding Bᵀ, i.e. logical B[k][j] = bmat[j*bstride+k].
//        (This is exactly W for out = in*Wᵀ, or any symmetric matrix.)
//        B-frag: lane l reads bmat row (j*16 + l&15), K slice (k0 + koff) -> 8-byte load.
__device__ __forceinline__ void gemm16x64(const float* __restrict__ arow,
                                          const float* __restrict__ bmat,
                                          int bstride, int koff, int l15,
                                          v8f (&c)[4]) {
#pragma unroll
  for (int k0 = 0; k0 < kD; k0 += 4) {
    v2f a = *(const v2f*)(arow + k0 + koff);
#pragma unroll
    for (int j = 0; j < 4; ++j) {
      v2f bf = *(const v2f*)(bmat + (j * 16 + l15) * bstride + k0 + koff);
      c[j] = wmma_f32(a, bf, c[j]);
    }
  }
}

__global__ void zero_kernel(float* __restrict__ p, int n) {
  int i = blockIdx.x * 256 + threadIdx.x;
  if (i < n) p[i] = 0.f;
}

// Stage 1: y = normalize(relu(relu(x W0ᵀ) W1ᵀ)) and G[b] += yᵀy (partial per block)
__global__ __launch_bounds__(kTPB) void encode_gram_kernel(
    const float* __restrict__ x, const float* __restrict__ Wenc,
    float* __restrict__ y, float* __restrict__ G) {
  __shared__ __align__(16) float sM[kD * kP];          // W0 -> W1 -> Gram scratch
  __shared__ __align__(16) float sH[kWAVES][16][kP];   // per-wave activation tile

  const int tid   = threadIdx.x;
  const int wave  = tid >> 5;
  const int lane  = tid & 31;
  const int l15   = lane & 15;
  const int koff  = (lane < 16) ? 0 : 2;
  const int rbase = (lane < 16) ? 0 : 8;  // C-layout: vgpr r -> row r / r+8
  const int b     = blockIdx.x >> 5;
  const long gtok0 = (long)b * kN + (blockIdx.x & 31) * kTokPerBlk + wave * 16;

  float (*h)[kP] = sH[wave];
  v8f zv = {};

  // ---- W_enc layer 0 -> LDS ----
  for (int i = tid; i < kD * kD; i += kTPB)
    sM[(i >> 6) * kP + (i & 63)] = Wenc[i];
  __syncthreads();

  // ---- layer 0: relu(x @ W0ᵀ), A-frags straight from global (b64, coalesced) ----
  v8f c[4] = {zv, zv, zv, zv};
  gemm16x64(x + (gtok0 + l15) * kD, sM, kP, koff, l15, c);
#pragma unroll
  for (int j = 0; j < 4; ++j)
#pragma unroll
    for (int r = 0; r < 8; ++r)
      h[rbase + r][j * 16 + l15] = fmaxf(c[j][r], 0.f);

  __syncthreads();
  for (int i = tid; i < kD * kD; i += kTPB)
    sM[(i >> 6) * kP + (i & 63)] = Wenc[kD * kD + i];
  __syncthreads();

  // ---- layer 1 + relu + L2 row-normalize ----
  v8f d[4] = {zv, zv, zv, zv};
  gemm16x64(&h[l15][0], sM, kP, koff, l15, d);

  float p[8];
#pragma unroll
  for (int r = 0; r < 8; ++r) {
    float s = 0.f;
#pragma unroll
    for (int j = 0; j < 4; ++j) {
      float v = fmaxf(d[j][r], 0.f);
      d[j][r] = v;
      s += v * v;
    }
    p[r] = s;
  }
  // row r (or r+8) lives entirely in one 16-lane half: butterfly within halves
#pragma unroll
  for (int m = 1; m <= 8; m <<= 1)
#pragma unroll
    for (int r = 0; r < 8; ++r) p[r] += __shfl_xor(p[r], m, 32);
#pragma unroll
  for (int r = 0; r < 8; ++r) p[r] = 1.f / (sqrtf(p[r]) + kEPS);

  float* yt = y + gtok0 * kD;
#pragma unroll
  for (int r = 0; r < 8; ++r) {
    int row = rbase + r;
#pragma unroll
    for (int j = 0; j < 4; ++j) {
      float v = d[j][r] * p[r];
      h[row][j * 16 + l15]    = v;   // keep tile for Gram
      yt[row * kD + j * 16 + l15] = v;
    }
  }

  // ---- Gram partial: G += yᵀ y over this wave's 16 tokens ----
  __syncthreads();                                        // done with sM as W1
  for (int i = tid; i < kD * kD; i += kTPB) sM[i] = 0.f;  // reuse as sG[64][64]
  __syncthreads();

  v8f g[16];
#pragma unroll
  for (int i = 0; i < 16; ++i) g[i] = zv;

#pragma unroll
  for (int t0 = 0; t0 < 16; t0 += 4) {
    // frag(q): A[d1,t]=y[t][16q+d1] and B[t,d2]=y[t][16q+d2] share one layout
    v2f f[4];
#pragma unroll
    for (int q = 0; q < 4; ++q) {
      v2f t;
      t.x = h[t0 + koff][q * 16 + l15];
      t.y = h[t0 + koff + 1][q * 16 + l15];
      f[q] = t;
    }
#pragma unroll
    for (int i = 0; i < 4; ++i)
#pragma unroll
      for (int j = 0; j < 4; ++j)
        g[i * 4 + j] = wmma_f32(f[i], f[j], g[i * 4 + j]);
  }

#pragma unroll
  for (int i = 0; i < 4; ++i)
#pragma unroll
    for (int j = 0; j < 4; ++j)
#pragma unroll
      for (int r = 0; r < 8; ++r)
        atomicAdd(&sM[(i * 16 + rbase + r) * kD + (j * 16 + l15)],
                  g[i * 4 + j][r]);                       // ds_add_f32
  __syncthreads();

  float* Gb = G + b * kD * kD;
  for (int i = tid; i < kD * kD; i += kTPB) atomicAdd(&Gb[i], sM[i]);
}

// Stage 2: out = relu(relu((y G) W0dᵀ) W1dᵀ)   (G symmetric)
__global__ __launch_bounds__(kTPB) void apply_decode_kernel(
    const float* __restrict__ y, const float* __restrict__ Wdec,
    const float* __restrict__ G, float* __restrict__ out) {
  __shared__ __align__(16) float sM[kD * kP];
  __shared__ __align__(16) float sH[kWAVES][16][kP];

  const int tid   = threadIdx.x;
  const int wave  = tid >> 5;
  const int lane  = tid & 31;
  const int l15   = lane & 15;
  const int koff  = (lane < 16) ? 0 : 2;
  const int rbase = (lane < 16) ? 0 : 8;
  const int b     = blockIdx.x >> 5;
  const long gtok0 = (long)b * kN + (blockIdx.x & 31) * kTokPerBlk + wave * 16;

  float (*h)[kP] = sH[wave];
  v8f zv = {};

  // ---- G[b] -> LDS (symmetric, so row-major works as gemm's Bᵀ form) ----
  const float* Gb = G + b * kD * kD;
  for (int i = tid; i < kD * kD; i += kTPB)
    sM[(i >> 6) * kP + (i & 63)] = Gb[i];
  __syncthreads();

  // ---- z = y @ G ----
  v8f c[4] = {zv, zv, zv, zv};
  gemm16x64(y + (gtok0 + l15) * kD, sM, kP, koff, l15, c);
#pragma unroll
  for (int j = 0; j < 4; ++j)
#pragma unroll
    for (int r = 0; r < 8; ++r)
      h[rbase + r][j * 16 + l15] = c[j][r];   // no relu here

  __syncthreads();
  for (int i = tid; i < kD * kD; i += kTPB)
    sM[(i >> 6) * kP + (i & 63)] = Wdec[i];
  __syncthreads();

  // ---- decode layer 0 ----
  v8f d[4] = {zv, zv, zv, zv};
  gemm16x64(&h[l15][0], sM, kP, koff, l15, d);
#pragma unroll
  for (int j = 0; j < 4; ++j)
#pragma unroll
    for (int r = 0; r < 8; ++r)
      h[rbase + r][j * 16 + l15] = fmaxf(d[j][r], 0.f);

  __syncthreads();
  for (int i = tid; i < kD * kD; i += kTPB)
    sM[(i >> 6) * kP + (i & 63)] = Wdec[kD * kD + i];
  __syncthreads();

  // ---- decode layer 1 -> out ----
  v8f e[4] = {zv, zv, zv, zv};
  gemm16x64(&h[l15][0], sM, kP, koff, l15, e);

  float* ot = out + gtok0 * kD;
#pragma unroll
  for (int r = 0; r < 8; ++r) {
    int row = rbase + r;
#pragma unroll
    for (int j = 0; j < 4; ++j)
      ot[row * kD + j * 16 + l15] = fmaxf(e[j][r], 0.f);
  }
}

extern "C" void kernel_launch(void* const* d_in, const int* in_sizes, int n_in,
                              void* d_out, int out_size, void* d_ws,
                              size_t ws_size, hipStream_t stream) {
  const float* x    = (const float*)d_in[0];  // [4,4096,64]
  const float* Wenc = (const float*)d_in[1];  // [2,64,64]
  const float* Wdec = (const float*)d_in[2];  // [2,64,64]
  float* out = (float*)d_out;                 // [4,4096,64]

  float* y = (float*)d_ws;                        // 4*4096*64 floats (4 MB)
  float* G = y + (size_t)kBATCH * kN * kD;        // 4*64*64 floats (64 KB)

  zero_kernel<<<(kBATCH * kD * kD + 255) / 256, 256, 0, stream>>>(
      G, kBATCH * kD * kD);
  encode_gram_kernel<<<kBlocks, kTPB, 0, stream>>>(x, Wenc, y, G);
  apply_decode_kernel<<<kBlocks, kTPB, 0, stream>>>(y, Wdec, G, out);
}